// Attn_61735859913284
// MI455X (gfx1250) — compile-verified
//
#include <hip/hip_runtime.h>
#include <hip/hip_bf16.h>

typedef __attribute__((ext_vector_type(2))) float v2f;
typedef __attribute__((ext_vector_type(4))) float v4f;
typedef __attribute__((ext_vector_type(8))) float v8f;

#define HDIM 1024
#define BATCH 32
#define SEQ 4096

// ---------------------------------------------------------------------------
// Phase 1: v[b,h] = sum_k hidden[b,k] * W[k,h]   (M=32, N=1024, K=1024)
// One wave per 16x16 output tile: 2 M-tiles x 64 N-tiles = 128 waves.
// Uses V_WMMA_F32_16X16X4_F32, K-loop of 256 iterations.
// ---------------------------------------------------------------------------
__global__ __launch_bounds__(128) void attn_v_gemm_wmma(
    const float* __restrict__ hidden,  // [32,1024] row-major (A)
    const float* __restrict__ W,       // [1024,1024] row-major, B[k,n]=W[k*H+n]
    float* __restrict__ v)             // [32,1024]
{
    const int wave = (blockIdx.x * blockDim.x + threadIdx.x) >> 5;  // 0..127
    const int lane = threadIdx.x & 31;
    const int tm   = wave & 1;     // 0..1
    const int tn   = wave >> 1;    // 0..63
    const int m0   = tm * 16;
    const int n0   = tn * 16;
    const int l15  = lane & 15;
    const int khalf = (lane >> 4) << 1;   // lanes 0-15 -> K+0,K+1 ; lanes 16-31 -> K+2,K+3

    const float* Arow = hidden + (size_t)(m0 + l15) * HDIM + khalf;
    const float* Bcol = W + (size_t)khalf * HDIM + (n0 + l15);

    v8f c = {};
    #pragma unroll 4
    for (int k = 0; k < HDIM; k += 4) {
        v2f a, b;
        a[0] = Arow[k + 0];                 // A[m, k+khalf]
        a[1] = Arow[k + 1];                 // A[m, k+khalf+1]
        b[0] = Bcol[(size_t)k * HDIM];      // B[k+khalf,   n]
        b[1] = Bcol[(size_t)(k + 1) * HDIM];// B[k+khalf+1, n]
        // D = A*B + C   (8 args: neg_a, A, neg_b, B, c_mod, C, reuse_a, reuse_b)
        c = __builtin_amdgcn_wmma_f32_16x16x4_f32(false, a, false, b,
                                                  (short)0, c, false, false);
    }

    // D layout: VGPR r -> row m0+r (lanes 0-15) / m0+8+r (lanes 16-31), col n0+l15
    const int rbase = m0 + ((lane >> 4) << 3);
    #pragma unroll
    for (int r = 0; r < 8; ++r)
        v[(size_t)(rbase + r) * HDIM + n0 + l15] = c[r];
}

// ---------------------------------------------------------------------------
// Phase 2: score[b,s] = enc[b,s,:] . v[b,:]
// One wave per (b,s) row; float4 loads -> 512B contiguous per wave-instr.
// enc (512 MB, single-use) is loaded NON-TEMPORALLY so it doesn't thrash the
// 192 MB L2; v (128 KB, reused by all 131072 rows) stays on the cached path.
// ---------------------------------------------------------------------------
__global__ __launch_bounds__(256) void attn_score_kernel(
    const float* __restrict__ enc,   // [32,4096,1024]
    const float* __restrict__ v,     // [32,1024] (L2-resident)
    float* __restrict__ out)         // [32,4096] raw scores
{
    const int row  = (blockIdx.x * 256 + threadIdx.x) >> 5;  // 0..131071
    const int lane = threadIdx.x & 31;
    const int b    = row >> 12;                              // row / 4096

    const v4f* erow = (const v4f*)(enc + (size_t)row * HDIM);
    const v4f* vrow = (const v4f*)(v + (size_t)b * HDIM);

    float acc = 0.0f;
    #pragma unroll
    for (int i = 0; i < 8; ++i) {
        v4f e = __builtin_nontemporal_load(&erow[i * 32 + lane]);  // TH=NT stream
        v4f w = vrow[i * 32 + lane];                               // cached (hot)
        acc = fmaf(e[0], w[0], acc);
        acc = fmaf(e[1], w[1], acc);
        acc = fmaf(e[2], w[2], acc);
        acc = fmaf(e[3], w[3], acc);
    }
    #pragma unroll
    for (int off = 16; off > 0; off >>= 1)
        acc += __shfl_xor(acc, off, 32);
    if (lane == 0) out[row] = acc;
}

// ---------------------------------------------------------------------------
// Phase 3: in-place softmax over s (4096) per batch. Bias term c[b] dropped:
// softmax is shift-invariant per row.
// ---------------------------------------------------------------------------
__global__ __launch_bounds__(1024) void attn_softmax_kernel(float* __restrict__ out)
{
    float* row = out + (size_t)blockIdx.x * SEQ;
    const int tid  = threadIdx.x;
    const int wid  = tid >> 5;
    const int lane = tid & 31;

    __shared__ float smax[32];
    __shared__ float ssum[32];

    float x[4];
    float m = -INFINITY;
    #pragma unroll
    for (int j = 0; j < 4; ++j) {
        x[j] = row[tid + j * 1024];
        m = fmaxf(m, x[j]);
    }
    #pragma unroll
    for (int off = 16; off > 0; off >>= 1)
        m = fmaxf(m, __shfl_xor(m, off, 32));
    if (lane == 0) smax[wid] = m;
    __syncthreads();
    if (wid == 0) {
        float t = smax[lane];
        #pragma unroll
        for (int off = 16; off > 0; off >>= 1)
            t = fmaxf(t, __shfl_xor(t, off, 32));
        if (lane == 0) smax[0] = t;
    }
    __syncthreads();
    m = smax[0];

    float s = 0.0f;
    #pragma unroll
    for (int j = 0; j < 4; ++j) {
        x[j] = __expf(x[j] - m);
        s += x[j];
    }
    #pragma unroll
    for (int off = 16; off > 0; off >>= 1)
        s += __shfl_xor(s, off, 32);
    if (lane == 0) ssum[wid] = s;
    __syncthreads();
    if (wid == 0) {
        float t = ssum[lane];
        #pragma unroll
        for (int off = 16; off > 0; off >>= 1)
            t += __shfl_xor(t, off, 32);
        if (lane == 0) ssum[0] = t;
    }
    __syncthreads();

    const float inv = 1.0f / ssum[0];
    #pragma unroll
    for (int j = 0; j < 4; ++j)
        row[tid + j * 1024] = x[j] * inv;
}

// ---------------------------------------------------------------------------
extern "C" void kernel_launch(void* const* d_in, const int* in_sizes, int n_in,
                              void* d_out, int out_size, void* d_ws, size_t ws_size,
                              hipStream_t stream)
{
    const float* hidden = (const float*)d_in[0];   // [32,1,1024]
    const float* enc    = (const float*)d_in[1];   // [32,4096,1024]
    const float* W      = (const float*)d_in[2];   // [1024,1024]
    // d_in[3] (bias) intentionally unused: softmax(x + c) == softmax(x)

    float* out = (float*)d_out;                    // [32,4096]
    float* v   = (float*)d_ws;                     // [32,1024] = 128 KB scratch

    // Phase 1: v = hidden @ W via WMMA (128 waves = 32 blocks x 4 waves)
    attn_v_gemm_wmma<<<32, 128, 0, stream>>>(hidden, W, v);
    // Phase 2: raw scores, streams enc once (bandwidth-bound, ~22 us at 23.3 TB/s)
    attn_score_kernel<<<(BATCH * SEQ * 32) / 256, 256, 0, stream>>>(enc, v, out);
    // Phase 3: per-batch softmax, in place
    attn_softmax_kernel<<<BATCH, 1024, 0, stream>>>(out);
}